// MultiHeadAttention_49374944035149
// MI455X (gfx1250) — compile-verified
//
#include <hip/hip_runtime.h>
#include <cstdint>

// ---------------------------------------------------------------------------
// MultiHeadAttention forward for MI455X (gfx1250), wave32, WMMA f32 path.
//   attn [B,H,S,S] and out [B,S,D] concatenated into d_out.
// Precision: fp32 end-to-end; matmuls via V_WMMA_F32_16X16X4_F32 so numerics
// match the fp32 reference. Pipeline is HBM-bound (~1 GB traffic dominated by
// the 256 MB attn tensor) => f32 matrix rate hides under the 23.3 TB/s floor.
// Data movement: GLOBAL_LOAD_ASYNC_TO_LDS_B128 (ASYNCcnt) + double-buffered
// LDS tiles so the async engine streams tile k+1 while WMMAs consume tile k.
// ---------------------------------------------------------------------------

typedef __attribute__((ext_vector_type(2))) float v2f;
typedef __attribute__((ext_vector_type(8))) float v8f;

#define HEADS   16
#define DK      64
#define DMOD    1024
#define SEQ     1024
#define BATCH   4
#define BS      (BATCH * SEQ)          // 4096 rows
#define NEG_INF_F (-1.0e9f)

#define TM 128      // block tile M
#define TN 64       // block tile N
#define TK 32       // K step
#define LDAP 36     // LDS pad (multiple of 4 -> 16B-aligned b128 lands)
#define LDBP 36
#define LDBNP 68    // NN B tile pad for 64-wide rows

// ----- gfx1250 async-to-LDS plumbing ---------------------------------------
#if defined(__gfx1250__)
#define GFX1250_DEV 1
#endif

#if defined(__has_builtin)
#if __has_builtin(__builtin_amdgcn_global_load_async_to_lds_b128)
#define HAVE_ASYNC_BUILTIN 1
#endif
#if __has_builtin(__builtin_amdgcn_s_wait_asynccnt)
#define HAVE_WAIT_ASYNC_BUILTIN 1
#endif
#endif

// Exact parameter type per hipcc diagnostic:
//   "__attribute__((__vector_size__(4 * sizeof(int)))) int __device__ *"
typedef int v4i_b128 __attribute__((vector_size(4 * sizeof(int))));
typedef __attribute__((address_space(1))) v4i_b128 gv4i;   // global (AS1)
typedef __attribute__((address_space(3))) v4i_b128 lv4i;   // LDS (AS3)
typedef __attribute__((address_space(3))) float lfloat;

// 16-byte global -> LDS copy. On gfx1250 this is the async engine
// (global_load_async_to_lds_b128, tracked by ASYNCcnt, no VGPR data path).
__device__ __forceinline__ void cp16_async(float* lds_dst, const float* gsrc) {
#if defined(HAVE_ASYNC_BUILTIN)
  __builtin_amdgcn_global_load_async_to_lds_b128(
      (gv4i*)gsrc, (lv4i*)lds_dst, /*offset=*/0, /*cpol=*/0);
#elif defined(GFX1250_DEV)
  // GV mode: LDS-dest VGPR (addrspace(3) ptr = 32-bit), 64-bit vaddr, no saddr
  asm volatile("global_load_async_to_lds_b128 %0, %1, off"
               :: "v"((lfloat*)lds_dst), "v"(gsrc)
               : "memory");
#else
  *reinterpret_cast<float4*>(lds_dst) =
      *reinterpret_cast<const float4*>(gsrc);
#endif
}

__device__ __forceinline__ void wait_async_all() {
#if defined(HAVE_WAIT_ASYNC_BUILTIN)
  __builtin_amdgcn_s_wait_asynccnt(0);
#elif defined(GFX1250_DEV)
  asm volatile("s_wait_asynccnt 0x0" ::: "memory");
#endif
}

__device__ __forceinline__ v8f wmma4(v2f a, v2f b, v8f c) {
  // D(16x16,f32) = A(16x4,f32) x B(4x16,f32) + C
  return __builtin_amdgcn_wmma_f32_16x16x4_f32(
      /*neg_a=*/false, a, /*neg_b=*/false, b,
      /*c_mod=*/(short)0, c, /*reuse_a=*/false, /*reuse_b=*/false);
}

// ---------------------------------------------------------------------------
// NT GEMM: C[M,N] = A[M,K] * B[N,K]^T   (dot rows of A with rows of B)
// EPI 0: projection  -> write [B,H,S,DK] permuted, scaled by outScale
// EPI 1: scores      -> per-(b,h) slab via blockIdx.z, mask epilogue
// EPI 2: out-proj    -> add residual, row-major [BS,DMOD]
// ---------------------------------------------------------------------------
template <int EPI>
__global__ __launch_bounds__(256) void gemm_nt(
    const float* __restrict__ Ag, const float* __restrict__ Bg,
    float* __restrict__ Cg, const int* __restrict__ maskg,
    const float* __restrict__ resid, int lda, int ldb, int K, float outScale) {
  __shared__ float As[2][TM][LDAP];
  __shared__ float Bs[2][TN][LDBP];

  const int t = threadIdx.x;
  const int blockM = blockIdx.x * TM;
  const int blockN = blockIdx.y * TN;

  const float* A = Ag;
  const float* B = Bg;
  float* C = Cg;
  const int* mrow = nullptr;
  if (EPI == 1) {
    const int bh = blockIdx.z;                       // b*16 + h
    A = Ag + (size_t)bh * SEQ * DK;                  // qh slab
    B = Bg + (size_t)bh * SEQ * DK;                  // kh slab (pre-scaled)
    C = Cg + (size_t)bh * SEQ * SEQ;                 // attn slab
    mrow = maskg + (size_t)(bh >> 4) * SEQ * SEQ;    // mask[b]
  }

  const int lane = t & 31;
  const int w    = t >> 5;          // 8 waves
  const int wm   = (w >> 1) << 5;   // wave M offset: 0..96
  const int wn   = (w & 1)  << 5;   // wave N offset: 0,32
  const int half = lane >> 4;       // lane group (0: lanes 0-15, 1: 16-31)
  const int l16  = lane & 15;

  v8f acc[2][2] = {};

  // async-stage one K tile into LDS buffer `buf`
  auto stage = [&](int buf, int k0) {
#pragma unroll
    for (int i = 0; i < 4; ++i) {                    // A: TMxTK, 4 b128/thread
      const int idx = t + (i << 8);
      const int r = idx >> 3;                        // 8 b128 per 32-float row
      const int c = (idx & 7) << 2;
      cp16_async(&As[buf][r][c], A + (size_t)(blockM + r) * lda + k0 + c);
    }
#pragma unroll
    for (int i = 0; i < 2; ++i) {                    // B: TNxTK, 2 b128/thread
      const int idx = t + (i << 8);
      const int r = idx >> 3;
      const int c = (idx & 7) << 2;
      cp16_async(&Bs[buf][r][c], B + (size_t)(blockN + r) * ldb + k0 + c);
    }
  };

  stage(0, 0);
  wait_async_all();
  __syncthreads();

  int cur = 0;
  for (int k0 = 0; k0 < K; k0 += TK) {
    const bool more = (k0 + TK) < K;
    if (more) stage(cur ^ 1, k0 + TK);   // async engine runs during WMMAs

#pragma unroll
    for (int kk = 0; kk < TK; kk += 4) {
      const int ka = kk + (half << 1);   // per-lane K base (ISA A/B layout)
      const v2f a0 = *reinterpret_cast<const v2f*>(&As[cur][wm + l16][ka]);
      const v2f a1 = *reinterpret_cast<const v2f*>(&As[cur][wm + 16 + l16][ka]);
      const v2f b0 = *reinterpret_cast<const v2f*>(&Bs[cur][wn + l16][ka]);
      const v2f b1 = *reinterpret_cast<const v2f*>(&Bs[cur][wn + 16 + l16][ka]);
      acc[0][0] = wmma4(a0, b0, acc[0][0]);
      acc[0][1] = wmma4(a0, b1, acc[0][1]);
      acc[1][0] = wmma4(a1, b0, acc[1][0]);
      acc[1][1] = wmma4(a1, b1, acc[1][1]);
    }
    if (more) {
      wait_async_all();                  // own wave's async writes done
      __syncthreads();                   // visible to all waves; reads drained
      cur ^= 1;
    }
  }

  // ---- epilogue: C layout per ISA (vgpr v -> M = v + 8*half, N = lane%16) --
#pragma unroll
  for (int mi = 0; mi < 2; ++mi) {
#pragma unroll
    for (int ni = 0; ni < 2; ++ni) {
#pragma unroll
      for (int v = 0; v < 8; ++v) {
        const int gm = blockM + wm + (mi << 4) + (half << 3) + v;
        const int gn = blockN + wn + (ni << 4) + l16;
        float val = acc[mi][ni][v];
        if (EPI == 0) {
          val *= outScale;
          const int b = gm >> 10, s = gm & 1023;
          const int h = gn >> 6,  d = gn & 63;
          C[((size_t)(b * HEADS + h) * SEQ + s) * DK + d] = val;
        } else if (EPI == 1) {
          if (mrow[(size_t)gm * SEQ + gn] == 0) val = NEG_INF_F;
          C[(size_t)gm * SEQ + gn] = val;
        } else {
          val += resid[(size_t)gm * DMOD + gn];
          C[(size_t)gm * DMOD + gn] = val;
        }
      }
    }
  }
}

// ---------------------------------------------------------------------------
// NN GEMM (context): ctx[i,d] = sum_j attn[i,j] * V[j,d], per (b,h) slab.
// Writes ctx in [B, S, H*DK] layout ready for the output projection.
// ---------------------------------------------------------------------------
__global__ __launch_bounds__(256) void gemm_nn_ctx(
    const float* __restrict__ attng, const float* __restrict__ vhg,
    float* __restrict__ ctx2) {
  __shared__ float As[2][TM][LDAP];
  __shared__ float Bs[2][TK][LDBNP];

  const int t = threadIdx.x;
  const int bh = blockIdx.z;
  const int blockM = blockIdx.x * TM;
  const float* A = attng + (size_t)bh * SEQ * SEQ;   // [i][j], lda = SEQ
  const float* B = vhg   + (size_t)bh * SEQ * DK;    // [j][d], ldb = DK
  const int bB = bh >> 4, hH = bh & 15;

  const int lane = t & 31;
  const int w    = t >> 5;
  const int wm   = (w >> 1) << 5;
  const int wn   = (w & 1)  << 5;
  const int half = lane >> 4;
  const int l16  = lane & 15;

  v8f acc[2][2] = {};

  auto stage = [&](int buf, int k0) {
#pragma unroll
    for (int i = 0; i < 4; ++i) {                    // A: TMxTK
      const int idx = t + (i << 8);
      const int r = idx >> 3;
      const int c = (idx & 7) << 2;
      cp16_async(&As[buf][r][c], A + (size_t)(blockM + r) * SEQ + k0 + c);
    }
#pragma unroll
    for (int i = 0; i < 2; ++i) {                    // B: TKx64 (k-major rows)
      const int idx = t + (i << 8);
      const int r = idx >> 4;                        // 16 b128 per 64-float row
      const int c = (idx & 15) << 2;
      cp16_async(&Bs[buf][r][c], B + (size_t)(k0 + r) * DK + c);
    }
  };

  stage(0, 0);
  wait_async_all();
  __syncthreads();

  int cur = 0;
  for (int k0 = 0; k0 < SEQ; k0 += TK) {
    const bool more = (k0 + TK) < SEQ;
    if (more) stage(cur ^ 1, k0 + TK);

#pragma unroll
    for (int kk = 0; kk < TK; kk += 4) {
      const int ka = kk + (half << 1);
      const v2f a0 = *reinterpret_cast<const v2f*>(&As[cur][wm + l16][ka]);
      const v2f a1 = *reinterpret_cast<const v2f*>(&As[cur][wm + 16 + l16][ka]);
      const v2f b0 = {Bs[cur][ka][wn + l16],      Bs[cur][ka + 1][wn + l16]};
      const v2f b1 = {Bs[cur][ka][wn + 16 + l16], Bs[cur][ka + 1][wn + 16 + l16]};
      acc[0][0] = wmma4(a0, b0, acc[0][0]);
      acc[0][1] = wmma4(a0, b1, acc[0][1]);
      acc[1][0] = wmma4(a1, b0, acc[1][0]);
      acc[1][1] = wmma4(a1, b1, acc[1][1]);
    }
    if (more) {
      wait_async_all();
      __syncthreads();
      cur ^= 1;
    }
  }

#pragma unroll
  for (int mi = 0; mi < 2; ++mi) {
#pragma unroll
    for (int ni = 0; ni < 2; ++ni) {
#pragma unroll
      for (int v = 0; v < 8; ++v) {
        const int gm = blockM + wm + (mi << 4) + (half << 3) + v;   // seq row
        const int gn = wn + (ni << 4) + l16;                        // d in [0,64)
        ctx2[((size_t)(bB * SEQ + gm)) * DMOD + hH * DK + gn] = acc[mi][ni][v];
      }
    }
  }
}

// ---------------------------------------------------------------------------
// Softmax: one wave32 per 1024-wide row, in place. shfl_xor reductions.
// ---------------------------------------------------------------------------
__global__ __launch_bounds__(256) void softmax_rows(float* __restrict__ attn) {
  const int row  = (blockIdx.x << 3) + (threadIdx.x >> 5);
  const int lane = threadIdx.x & 31;
  float* p = attn + (size_t)row * SEQ;

  float vals[32];
  float m = -3.0e9f;
#pragma unroll
  for (int i = 0; i < 32; ++i) {
    vals[i] = p[lane + (i << 5)];
    m = fmaxf(m, vals[i]);
  }
#pragma unroll
  for (int off = 16; off; off >>= 1) m = fmaxf(m, __shfl_xor(m, off, 32));

  float s = 0.f;
#pragma unroll
  for (int i = 0; i < 32; ++i) {
    const float e = __expf(vals[i] - m);
    vals[i] = e;
    s += e;
  }
#pragma unroll
  for (int off = 16; off; off >>= 1) s += __shfl_xor(s, off, 32);

  const float inv = 1.0f / s;
#pragma unroll
  for (int i = 0; i < 32; ++i) p[lane + (i << 5)] = vals[i] * inv;
}

// ---------------------------------------------------------------------------
// LayerNorm: one 256-thread block per 1024-wide row.
// ---------------------------------------------------------------------------
__global__ __launch_bounds__(256) void layernorm_rows(
    const float* __restrict__ x, const float* __restrict__ gamma,
    const float* __restrict__ beta, float* __restrict__ out) {
  __shared__ float red[2][8];
  const int row = blockIdx.x;
  const int t = threadIdx.x;
  const float* p = x + (size_t)row * DMOD;

  float v[4];
  float s = 0.f, ss = 0.f;
#pragma unroll
  for (int i = 0; i < 4; ++i) {
    v[i] = p[t + (i << 8)];
    s += v[i];
    ss += v[i] * v[i];
  }
#pragma unroll
  for (int off = 16; off; off >>= 1) {
    s  += __shfl_xor(s, off, 32);
    ss += __shfl_xor(ss, off, 32);
  }
  const int w = t >> 5, lane = t & 31;
  if (lane == 0) { red[0][w] = s; red[1][w] = ss; }
  __syncthreads();
  s = 0.f; ss = 0.f;
#pragma unroll
  for (int i = 0; i < 8; ++i) { s += red[0][i]; ss += red[1][i]; }

  const float mu  = s * (1.0f / DMOD);
  const float var = ss * (1.0f / DMOD) - mu * mu;
  const float r   = rsqrtf(var + 1e-6f);

  float* o = out + (size_t)row * DMOD;
#pragma unroll
  for (int i = 0; i < 4; ++i) {
    const int c = t + (i << 8);
    o[c] = (v[i] - mu) * r * gamma[c] + beta[c];
  }
}

// ---------------------------------------------------------------------------
extern "C" void kernel_launch(void* const* d_in, const int* in_sizes, int n_in,
                              void* d_out, int out_size, void* d_ws,
                              size_t ws_size, hipStream_t stream) {
  const float* q     = (const float*)d_in[0];
  const float* k     = (const float*)d_in[1];
  const float* v     = (const float*)d_in[2];
  const int*   mask  = (const int*)  d_in[3];
  const float* Wq    = (const float*)d_in[4];
  const float* Wk    = (const float*)d_in[5];
  const float* Wv    = (const float*)d_in[6];
  const float* Wo    = (const float*)d_in[7];
  const float* gamma = (const float*)d_in[8];
  const float* beta  = (const float*)d_in[9];

  float* attn = (float*)d_out;                                   // [B,H,S,S]
  float* outp = (float*)d_out + (size_t)BATCH * HEADS * SEQ * SEQ;

  const size_t PROJ = (size_t)BATCH * HEADS * SEQ * DK;          // 4M floats
  float* ws    = (float*)d_ws;
  float* qh    = ws;                 // [B,H,S,DK]
  float* kh    = ws + PROJ;          // pre-scaled by 1/sqrt(DK)
  float* vh    = ws + 2 * PROJ;
  float* ctx2  = ws + 3 * PROJ;      // [B,S,H*DK]
  float* preLN = ws;                 // reuse qh region (safe: qh dead by then)

  const dim3 blk(256);

  // QKV projections: [4096,1024] x W^T
  const dim3 gProj(BS / TM, DMOD / TN);
  gemm_nt<0><<<gProj, blk, 0, stream>>>(q, Wq, qh, nullptr, nullptr,
                                        DMOD, DMOD, DMOD, 1.0f);
  gemm_nt<0><<<gProj, blk, 0, stream>>>(k, Wk, kh, nullptr, nullptr,
                                        DMOD, DMOD, DMOD, 0.125f);  // 1/sqrt(64)
  gemm_nt<0><<<gProj, blk, 0, stream>>>(v, Wv, vh, nullptr, nullptr,
                                        DMOD, DMOD, DMOD, 1.0f);

  // Scores + mask -> attn region of d_out (64 slabs)
  const dim3 gSc(SEQ / TM, SEQ / TN, BATCH * HEADS);
  gemm_nt<1><<<gSc, blk, 0, stream>>>(qh, kh, attn, mask, nullptr,
                                      DK, DK, DK, 1.0f);

  // Softmax in place (8 rows per block, one wave each)
  softmax_rows<<<dim3(BATCH * HEADS * SEQ / 8), blk, 0, stream>>>(attn);

  // Context: attn x V -> ctx2 [B,S,H*DK]
  gemm_nn_ctx<<<dim3(SEQ / TM, 1, BATCH * HEADS), blk, 0, stream>>>(attn, vh, ctx2);

  // Output projection + residual -> preLN
  gemm_nt<2><<<gProj, blk, 0, stream>>>(ctx2, Wo, preLN, nullptr, q,
                                        DMOD, DMOD, DMOD, 1.0f);

  // LayerNorm -> out region of d_out
  layernorm_rows<<<dim3(BS), blk, 0, stream>>>(preLN, gamma, beta, outp);
}